// RootCauseGNN_35588099014823
// MI455X (gfx1250) — compile-verified
//
#include <hip/hip_runtime.h>
#include <hip/hip_bf16.h>
#include <stdint.h>

// -----------------------------------------------------------------------------
// 3-layer GAT for gfx1250 (MI455X).
//  - GEMMs (x@W) via V_WMMA_F32_16X16X4_F32 (exact f32, matches reference).
//    Each wave computes TWO 16x16 row-tiles sharing one B fetch (2x WMMA per
//    load); A-tile staged in LDS with +4-float row padding (conflict-free
//    ds_load on 64 banks); staging uses float4 (b128) loads.
//  - Edge softmax via native atomics: global_atomic_max_u32 on an
//    order-preserving uint encoding of f32, global_atomic_add_f32 for sums.
//  - Aggregation: one wave32 per (edge, head), lane == channel (out_c == 32).
//  - Working set (~60 MB) fits MI455X's 192 MB L2, so edge passes run at L2 BW.
// -----------------------------------------------------------------------------

typedef float v2f __attribute__((ext_vector_type(2)));
typedef float v8f __attribute__((ext_vector_type(8)));

#define NEG_SLOPE 0.2f
#define ENC_NEG_INF 0x007FFFFFu  // ordKey(-inf)

__device__ __forceinline__ unsigned ordKey(float f) {
  unsigned b = __float_as_uint(f);
  return (b & 0x80000000u) ? ~b : (b | 0x80000000u);
}
__device__ __forceinline__ float ordDec(unsigned k) {
  unsigned b = (k & 0x80000000u) ? (k & 0x7FFFFFFFu) : ~k;
  return __uint_as_float(b);
}
__device__ __forceinline__ float leaky(float l) {
  return l > 0.f ? l : NEG_SLOPE * l;
}

// ---------------------------------------------------------------------------
// GEMM: H[n,M] = X[n,K] @ W[K,M], f32, WMMA 16x16x4.
// Block = 256 threads = 8 waves. Wave (wrt, ct) computes row-tiles
// {2*wrt, 2*wrt+1} x col-tile ct, sharing one B fetch per k-step.
// A layout (16x4 f32): lane<16 -> M=lane, VGPR{0,1}=K{0,1}; lane>=16 -> K{2,3}.
// B layout (4x16 f32): VGPR j -> K=j (lanes 0-15), K=j+2 (lanes 16-31).
// C layout: VGPR i -> M = i (+8 for lanes 16-31), N = lane&15.
// ---------------------------------------------------------------------------
template <int K, int M>
__global__ __launch_bounds__(256) void gemm_wmma_f32(
    const float* __restrict__ X, const float* __restrict__ W,
    float* __restrict__ H, int nrows) {
  constexpr int COL_TILES = M / 16;
  constexpr int ROW_TILES = (8 / COL_TILES) * 2;  // 2 row-tiles per wave
  constexpr int ROWS = ROW_TILES * 16;
  constexpr int LROW = K + 4;  // +4 floats: lanes land 4 banks apart
  __shared__ float xs[ROWS * LROW];

  const int row_base = blockIdx.x * ROWS;

  // Stage X tile as float4 (global_load_b128 / ds_store_b128).
  constexpr int KV = K / 4;
  for (int i = threadIdx.x; i < ROWS * KV; i += 256) {
    int r = i / KV, c4 = i % KV;
    int gr = row_base + r;
    float4 v = (gr < nrows) ? ((const float4*)X)[(size_t)gr * KV + c4]
                            : make_float4(0.f, 0.f, 0.f, 0.f);
    *(float4*)&xs[r * LROW + c4 * 4] = v;
  }
  __syncthreads();

  const int wave = threadIdx.x >> 5, lane = threadIdx.x & 31;
  const int wrt = wave / COL_TILES, ct = wave % COL_TILES;
  const int lm = lane & 15, half = lane >> 4;
  const float* xrow0 = &xs[(wrt * 32 + lm) * LROW];
  const float* xrow1 = xrow0 + 16 * LROW;
  const int col = ct * 16 + lm;

  v8f acc0 = {0.f, 0.f, 0.f, 0.f, 0.f, 0.f, 0.f, 0.f};
  v8f acc1 = {0.f, 0.f, 0.f, 0.f, 0.f, 0.f, 0.f, 0.f};
#pragma unroll
  for (int k0 = 0; k0 < K; k0 += 4) {
    const int ka = k0 + 2 * half;
    v2f a0, a1, b;
    a0.x = xrow0[ka];
    a0.y = xrow0[ka + 1];
    a1.x = xrow1[ka];
    a1.y = xrow1[ka + 1];
    b.x = W[(size_t)ka * M + col];
    b.y = W[(size_t)(ka + 1) * M + col];
    acc0 = __builtin_amdgcn_wmma_f32_16x16x4_f32(false, a0, false, b, (short)0,
                                                 acc0, false, false);
    acc1 = __builtin_amdgcn_wmma_f32_16x16x4_f32(false, a1, false, b, (short)0,
                                                 acc1, false, false);
  }

  const int out_row0 = row_base + wrt * 32 + 8 * half;
#pragma unroll
  for (int i = 0; i < 8; ++i) {
    int r0 = out_row0 + i;
    if (r0 < nrows) H[(size_t)r0 * M + col] = acc0[i];
    int r1 = out_row0 + 16 + i;
    if (r1 < nrows) H[(size_t)r1 * M + col] = acc1[i];
  }
}

// ---------------------------------------------------------------------------
// Per-node attention logits: als[n,h] = <h[n,h,:], a_src[h,:]>, same for dst.
// ---------------------------------------------------------------------------
__global__ void node_logits(const float* __restrict__ h,
                            const float* __restrict__ a_src,
                            const float* __restrict__ a_dst,
                            float* __restrict__ als, float* __restrict__ ald,
                            int n, int heads, int outc) {
  int t = blockIdx.x * blockDim.x + threadIdx.x;
  if (t >= n * heads) return;
  int node = t / heads, hd = t % heads;
  const float* hp = h + ((size_t)node * heads + hd) * outc;
  float s = 0.f, d = 0.f;
  for (int c = 0; c < outc; ++c) {
    float v = hp[c];
    s += v * a_src[hd * outc + c];
    d += v * a_dst[hd * outc + c];
  }
  als[t] = s;
  ald[t] = d;
}

__global__ void fill_f32(float* __restrict__ p, float v, int count) {
  int t = blockIdx.x * blockDim.x + threadIdx.x;
  if (t < count) p[t] = v;
}
__global__ void fill_u32(unsigned* __restrict__ p, unsigned v, int count) {
  int t = blockIdx.x * blockDim.x + threadIdx.x;
  if (t < count) p[t] = v;
}

__device__ __forceinline__ void edge_ends(const int* __restrict__ ei, int E,
                                          int e, int& s, int& d) {
  if (e < E) {
    s = ei[e];
    d = ei[E + e];
  } else {  // self-loop
    s = d = e - E;
  }
}

// ---- segment max over incoming edges (order-preserving uint atomics) ------
__global__ void edge_softmax_max(const int* __restrict__ ei, int E, int n,
                                 int heads, const float* __restrict__ als,
                                 const float* __restrict__ ald,
                                 unsigned* __restrict__ mx) {
  long long t = (long long)blockIdx.x * blockDim.x + threadIdx.x;
  long long tot = (long long)(E + n) * heads;
  if (t >= tot) return;
  int e = (int)(t / heads), hd = (int)(t % heads);
  int s, d;
  edge_ends(ei, E, e, s, d);
  float l = leaky(als[s * heads + hd] + ald[d * heads + hd]);
  atomicMax(&mx[d * heads + hd], ordKey(l));
}

// ---- segment sum of exp(l - m) --------------------------------------------
__global__ void edge_softmax_sum(const int* __restrict__ ei, int E, int n,
                                 int heads, const float* __restrict__ als,
                                 const float* __restrict__ ald,
                                 const unsigned* __restrict__ mx,
                                 float* __restrict__ den) {
  long long t = (long long)blockIdx.x * blockDim.x + threadIdx.x;
  long long tot = (long long)(E + n) * heads;
  if (t >= tot) return;
  int e = (int)(t / heads), hd = (int)(t % heads);
  int s, d;
  edge_ends(ei, E, e, s, d);
  float l = leaky(als[s * heads + hd] + ald[d * heads + hd]);
  float m = ordDec(mx[d * heads + hd]);
  atomicAdd(&den[d * heads + hd], __expf(l - m));
}

// ---- weighted scatter-add, out_c==32 : one wave per (edge, head) ----------
__global__ void edge_aggregate32(const int* __restrict__ ei, int E, int n,
                                 int heads, const float* __restrict__ als,
                                 const float* __restrict__ ald,
                                 const unsigned* __restrict__ mx,
                                 const float* __restrict__ den,
                                 const float* __restrict__ h,
                                 float* __restrict__ agg) {
  long long gt = (long long)blockIdx.x * blockDim.x + threadIdx.x;
  long long w = gt >> 5;
  int lane = (int)(gt & 31);
  long long tot = (long long)(E + n) * heads;
  if (w >= tot) return;
  int e = (int)(w / heads), hd = (int)(w % heads);
  int s, d;
  edge_ends(ei, E, e, s, d);
  float l = leaky(als[s * heads + hd] + ald[d * heads + hd]);
  float m = ordDec(mx[d * heads + hd]);
  float alpha = __expf(l - m) / den[d * heads + hd];
  const int F = heads * 32;
  atomicAdd(&agg[(size_t)d * F + hd * 32 + lane],
            alpha * h[(size_t)s * F + hd * 32 + lane]);
}

// ---- out_c==1 variant (layer 3), one thread per edge ----------------------
__global__ void edge_aggregate1(const int* __restrict__ ei, int E, int n,
                                const float* __restrict__ als,
                                const float* __restrict__ ald,
                                const unsigned* __restrict__ mx,
                                const float* __restrict__ den,
                                const float* __restrict__ h3,
                                float* __restrict__ agg) {
  int e = blockIdx.x * blockDim.x + threadIdx.x;
  if (e >= E + n) return;
  int s, d;
  edge_ends(ei, E, e, s, d);
  float l = leaky(als[s] + ald[d]);
  float alpha = __expf(l - ordDec(mx[d])) / den[d];
  atomicAdd(&agg[d], alpha * h3[s]);
}

__global__ void bias_act(float* __restrict__ x, const float* __restrict__ b,
                         long long count, int F, int do_elu) {
  long long t = (long long)blockIdx.x * blockDim.x + threadIdx.x;
  if (t >= count) return;
  float v = x[t] + b[(int)(t % F)];
  if (do_elu) v = v > 0.f ? v : (__expf(v) - 1.f);
  x[t] = v;
}

// layer-3 projection to 1 channel + its (1x1) attention logits, fused
__global__ void layer3_node(const float* __restrict__ h2, /* [n,64] */
                            const float* __restrict__ W3, /* [64,1] */
                            const float* __restrict__ a_s,
                            const float* __restrict__ a_d,
                            float* __restrict__ h3, float* __restrict__ als,
                            float* __restrict__ ald, int n) {
  int t = blockIdx.x * blockDim.x + threadIdx.x;
  if (t >= n) return;
  float s = 0.f;
  const float* row = h2 + (size_t)t * 64;
  for (int c = 0; c < 64; ++c) s += row[c] * W3[c];
  h3[t] = s;
  als[t] = s * a_s[0];
  ald[t] = s * a_d[0];
}

__global__ void add_scalar(float* __restrict__ p, const float* __restrict__ b,
                           int count) {
  int t = blockIdx.x * blockDim.x + threadIdx.x;
  if (t < count) p[t] += b[0];
}

// ---------------------------------------------------------------------------
extern "C" void kernel_launch(void* const* d_in, const int* in_sizes, int n_in,
                              void* d_out, int out_size, void* d_ws,
                              size_t ws_size, hipStream_t stream) {
  const float* x   = (const float*)d_in[0];
  const int*   ei  = (const int*)d_in[1];  // [2,E] row-major, int32
  const float* W1  = (const float*)d_in[2];
  const float* as1 = (const float*)d_in[3];
  const float* ad1 = (const float*)d_in[4];
  const float* b1  = (const float*)d_in[5];
  const float* W2  = (const float*)d_in[6];
  const float* as2 = (const float*)d_in[7];
  const float* ad2 = (const float*)d_in[8];
  const float* b2  = (const float*)d_in[9];
  const float* W3  = (const float*)d_in[10];
  const float* as3 = (const float*)d_in[11];
  const float* ad3 = (const float*)d_in[12];
  const float* b3  = (const float*)d_in[13];

  const int n = in_sizes[0] / 128;  // 50000
  const int E = in_sizes[1] / 2;    // 800000
  const int Etot = E + n;           // edges + self loops

  // Workspace layout (floats): ~n*273 floats ≈ 55 MB.
  float* H   = (float*)d_ws;              // n*128  (h1; later reused for h2)
  float* OUT = H + (size_t)n * 128;       // n*128  (out1; later agg2/out2)
  float* ALS = OUT + (size_t)n * 128;     // n*4
  float* ALD = ALS + (size_t)n * 4;       // n*4
  unsigned* MX = (unsigned*)(ALD + (size_t)n * 4);  // n*4
  float* DEN = (float*)(MX + (size_t)n * 4);        // n*4
  float* H3  = DEN + (size_t)n * 4;                 // n

  auto blocks = [](long long work, int tpb) {
    return dim3((unsigned)((work + tpb - 1) / tpb));
  };

  // ---------------- Layer 1: heads=4, out_c=32, M=128 ----------------
  // gemm<128,128>: ROWS = 32 per block
  gemm_wmma_f32<128, 128><<<(n + 31) / 32, 256, 0, stream>>>(x, W1, H, n);
  node_logits<<<blocks((long long)n * 4, 256), 256, 0, stream>>>(
      H, as1, ad1, ALS, ALD, n, 4, 32);
  fill_u32<<<blocks(n * 4, 256), 256, 0, stream>>>(MX, ENC_NEG_INF, n * 4);
  fill_f32<<<blocks(n * 4, 256), 256, 0, stream>>>(DEN, 0.f, n * 4);
  fill_f32<<<blocks((long long)n * 128, 256), 256, 0, stream>>>(OUT, 0.f,
                                                                n * 128);
  edge_softmax_max<<<blocks((long long)Etot * 4, 256), 256, 0, stream>>>(
      ei, E, n, 4, ALS, ALD, MX);
  edge_softmax_sum<<<blocks((long long)Etot * 4, 256), 256, 0, stream>>>(
      ei, E, n, 4, ALS, ALD, MX, DEN);
  edge_aggregate32<<<blocks((long long)Etot * 4 * 32, 256), 256, 0, stream>>>(
      ei, E, n, 4, ALS, ALD, MX, DEN, H, OUT);
  bias_act<<<blocks((long long)n * 128, 256), 256, 0, stream>>>(
      OUT, b1, (long long)n * 128, 128, 1);

  // ---------------- Layer 2: heads=2, out_c=32, M=64 -----------------
  // gemm<128,64>: ROWS = 64 per block
  gemm_wmma_f32<128, 64><<<(n + 63) / 64, 256, 0, stream>>>(OUT, W2, H, n);
  node_logits<<<blocks((long long)n * 2, 256), 256, 0, stream>>>(
      H, as2, ad2, ALS, ALD, n, 2, 32);
  fill_u32<<<blocks(n * 2, 256), 256, 0, stream>>>(MX, ENC_NEG_INF, n * 2);
  fill_f32<<<blocks(n * 2, 256), 256, 0, stream>>>(DEN, 0.f, n * 2);
  fill_f32<<<blocks((long long)n * 64, 256), 256, 0, stream>>>(OUT, 0.f,
                                                               n * 64);
  edge_softmax_max<<<blocks((long long)Etot * 2, 256), 256, 0, stream>>>(
      ei, E, n, 2, ALS, ALD, MX);
  edge_softmax_sum<<<blocks((long long)Etot * 2, 256), 256, 0, stream>>>(
      ei, E, n, 2, ALS, ALD, MX, DEN);
  edge_aggregate32<<<blocks((long long)Etot * 2 * 32, 256), 256, 0, stream>>>(
      ei, E, n, 2, ALS, ALD, MX, DEN, H, OUT);
  bias_act<<<blocks((long long)n * 64, 256), 256, 0, stream>>>(
      OUT, b2, (long long)n * 64, 64, 1);

  // ---------------- Layer 3: heads=1, out_c=1 ------------------------
  layer3_node<<<blocks(n, 256), 256, 0, stream>>>(OUT, W3, as3, ad3, H3, ALS,
                                                  ALD, n);
  fill_u32<<<blocks(n, 256), 256, 0, stream>>>(MX, ENC_NEG_INF, n);
  fill_f32<<<blocks(n, 256), 256, 0, stream>>>(DEN, 0.f, n);
  fill_f32<<<blocks(n, 256), 256, 0, stream>>>((float*)d_out, 0.f, n);
  edge_softmax_max<<<blocks(Etot, 256), 256, 0, stream>>>(ei, E, n, 1, ALS,
                                                          ALD, MX);
  edge_softmax_sum<<<blocks(Etot, 256), 256, 0, stream>>>(ei, E, n, 1, ALS,
                                                          ALD, MX, DEN);
  edge_aggregate1<<<blocks(Etot, 256), 256, 0, stream>>>(
      ei, E, n, ALS, ALD, MX, DEN, H3, (float*)d_out);
  add_scalar<<<blocks(n, 256), 256, 0, stream>>>((float*)d_out, b3, n);
}